// AdaFilter_1_26182120637032
// MI455X (gfx1250) — compile-verified
//
#include <hip/hip_runtime.h>

typedef __attribute__((ext_vector_type(2))) float v2f;
typedef __attribute__((ext_vector_type(8))) float v8f;

#define HH 192
#define WW 408
#define HWIMG (HH * WW) /* 78336 */
#define FS_STRIDE 340
#define G1S 404

__device__ __forceinline__ int refl(int i, int n) {
  if (i < 0) i = -i;
  if (i >= n) i = 2 * n - 2 - i;
  return i;
}
__device__ __forceinline__ float leaky(float v) { return v > 0.f ? v : 0.01f * v; }
__device__ __forceinline__ float waveRed(float v) {
#pragma unroll
  for (int o = 16; o > 0; o >>= 1) v += __shfl_down(v, o, 32);
  return v;
}

// ---------------- utility ----------------
__global__ void k_zero(float* p, int n) {
  int i = blockIdx.x * blockDim.x + threadIdx.x;
  if (i < n) p[i] = 0.f;
}

// ---------------- encoder stage 1: grouped conv 1->2/ch + pool(1,2) + GN stats ----------------
__global__ void k_enc1(const float* __restrict__ x, const float* __restrict__ w,
                       const float* __restrict__ bias, float* __restrict__ out,
                       float* __restrict__ gsum, float* __restrict__ gsq) {
  const int bid = blockIdx.x;  // n*64 + c
  const int n = bid >> 6, c = bid & 63, p = c >> 1;
  const float* xin = x + (size_t)(n * 32 + p) * HWIMG;
  float wr[9];
#pragma unroll
  for (int k = 0; k < 9; ++k) wr[k] = w[c * 9 + k];
  const float bv = bias[c];
  float* orow = out + (size_t)bid * (190 * 203);
  float lsum = 0.f, lsq = 0.f;
  for (int i = threadIdx.x; i < 190 * 203; i += blockDim.x) {
    int y = i / 203, xp = i % 203;
    float s = 0.f;
#pragma unroll
    for (int dx2 = 0; dx2 < 2; ++dx2) {
      int xb = 2 * xp + dx2;
#pragma unroll
      for (int dy = 0; dy < 3; ++dy)
#pragma unroll
        for (int dx = 0; dx < 3; ++dx)
          s += wr[dy * 3 + dx] * xin[(y + dy) * WW + xb + dx];
    }
    float v = 0.5f * s + bv;
    orow[i] = v;
    lsum += v;
    lsq += v * v;
  }
  __shared__ float accS, accQ;
  if (threadIdx.x == 0) { accS = 0.f; accQ = 0.f; }
  __syncthreads();
  float rs = waveRed(lsum), rq = waveRed(lsq);
  if ((threadIdx.x & 31) == 0) { atomicAdd(&accS, rs); atomicAdd(&accQ, rq); }
  __syncthreads();
  if (threadIdx.x == 0) {
    atomicAdd(&gsum[n * 32 + p], accS);
    atomicAdd(&gsq[n * 32 + p], accQ);
  }
}

// per-channel GN scale/shift from per-(n,group) sums
__global__ void k_gnfin(const float* gsum, const float* gsq, const float* g,
                        const float* b, float* scale, float* shift, int nch,
                        int chPerGrp, float cnt) {
  int t = blockIdx.x * blockDim.x + threadIdx.x;  // n*nch + c
  if (t >= 2 * nch) return;
  int n = t / nch, c = t % nch, p = c / chPerGrp;
  float mu = gsum[n * 32 + p] / cnt;
  float var = gsq[n * 32 + p] / cnt - mu * mu;
  float r = rsqrtf(var + 1e-5f);
  float sc = r * g[c];
  scale[t] = sc;
  shift[t] = b[c] - mu * sc;
}

// ---------------- encoder stage 2 ----------------
__global__ void k_enc2(const float* __restrict__ in, const float* __restrict__ w,
                       const float* __restrict__ bias, const float* __restrict__ scA,
                       const float* __restrict__ shA, float* __restrict__ out,
                       float* __restrict__ gsum, float* __restrict__ gsq) {
  const int bid = blockIdx.x;  // n*128 + c
  const int n = bid >> 7, c = bid & 127, p = c >> 2;
  const int icb = n * 64 + p * 2;
  const float* i0 = in + (size_t)icb * (190 * 203);
  const float* i1 = i0 + 190 * 203;
  const float s0 = scA[icb], h0 = shA[icb], s1 = scA[icb + 1], h1 = shA[icb + 1];
  float wr[18];
#pragma unroll
  for (int k = 0; k < 18; ++k) wr[k] = w[c * 18 + k];
  const float bv = bias[c];
  float* orow = out + (size_t)bid * (188 * 100);
  float lsum = 0.f, lsq = 0.f;
  for (int i = threadIdx.x; i < 188 * 100; i += blockDim.x) {
    int y = i / 100, xp = i % 100;
    float s = 0.f;
#pragma unroll
    for (int dx2 = 0; dx2 < 2; ++dx2) {
      int xb = 2 * xp + dx2;
#pragma unroll
      for (int dy = 0; dy < 3; ++dy)
#pragma unroll
        for (int dx = 0; dx < 3; ++dx) {
          int idx = (y + dy) * 203 + xb + dx;
          float a0 = fmaxf(i0[idx] * s0 + h0, 0.f);
          float a1 = fmaxf(i1[idx] * s1 + h1, 0.f);
          s += wr[dy * 3 + dx] * a0 + wr[9 + dy * 3 + dx] * a1;
        }
    }
    float v = 0.5f * s + bv;
    orow[i] = v;
    lsum += v;
    lsq += v * v;
  }
  __shared__ float accS, accQ;
  if (threadIdx.x == 0) { accS = 0.f; accQ = 0.f; }
  __syncthreads();
  float rs = waveRed(lsum), rq = waveRed(lsq);
  if ((threadIdx.x & 31) == 0) { atomicAdd(&accS, rs); atomicAdd(&accQ, rq); }
  __syncthreads();
  if (threadIdx.x == 0) {
    atomicAdd(&gsum[n * 32 + p], accS);
    atomicAdd(&gsq[n * 32 + p], accQ);
  }
}

// ---------------- encoder stage 3: conv + full 185x97 avg pool ----------------
__global__ void k_enc3(const float* __restrict__ in, const float* __restrict__ w,
                       const float* __restrict__ bias, const float* __restrict__ scB,
                       const float* __restrict__ shB, float* __restrict__ pooled) {
  const int bid = blockIdx.x;  // n*256 + c
  const int n = bid >> 8, c = bid & 255, p = c >> 3;
  const int icb = n * 128 + p * 4;
  const float* ip0 = in + (size_t)icb * (188 * 100);
  float sc[4], sh[4];
#pragma unroll
  for (int ic = 0; ic < 4; ++ic) { sc[ic] = scB[icb + ic]; sh[ic] = shB[icb + ic]; }
  float wr[36];
#pragma unroll
  for (int k = 0; k < 36; ++k) wr[k] = w[c * 36 + k];
  const float bv = bias[c];
  float lsum = 0.f;
  for (int i = threadIdx.x; i < 185 * 97; i += blockDim.x) {
    int y = i / 97, xx = i % 97;
    float s = bv;
#pragma unroll
    for (int ic = 0; ic < 4; ++ic) {
      const float* ir = ip0 + (size_t)ic * (188 * 100);
#pragma unroll
      for (int dy = 0; dy < 3; ++dy)
#pragma unroll
        for (int dx = 0; dx < 3; ++dx) {
          float a = fmaxf(ir[(y + dy) * 100 + xx + dx] * sc[ic] + sh[ic], 0.f);
          s += wr[ic * 9 + dy * 3 + dx] * a;
        }
    }
    lsum += s;
  }
  __shared__ float acc;
  if (threadIdx.x == 0) acc = 0.f;
  __syncthreads();
  float r = waveRed(lsum);
  if ((threadIdx.x & 31) == 0) atomicAdd(&acc, r);
  __syncthreads();
  if (threadIdx.x == 0) pooled[bid] = acc / (185.f * 97.f);
}

// GN3 over 8-channel groups on 512 pooled values -> feats
__global__ void k_gn3feats(const float* __restrict__ pooled, const float* __restrict__ g,
                           const float* __restrict__ b, float* __restrict__ feats) {
  __shared__ float sv[512];
  int t = threadIdx.x;
  sv[t] = pooled[t];
  __syncthreads();
  int base = t & ~7;
  float mu = 0.f;
#pragma unroll
  for (int j = 0; j < 8; ++j) mu += sv[base + j];
  mu *= 0.125f;
  float var = 0.f;
#pragma unroll
  for (int j = 0; j < 8; ++j) { float d = sv[base + j] - mu; var += d * d; }
  var *= 0.125f;
  int c = t & 255;
  float v = (sv[t] - mu) * rsqrtf(var + 1e-5f) * g[c] + b[c];
  feats[t] = fmaxf(v, 0.f);
}

// ---------------- tiny MLP predicting 305 filter params per (n,panel) ----------------
__global__ void k_mlp(const float* __restrict__ feats, const float* __restrict__ w1,
                      const float* __restrict__ b1, const float* __restrict__ w2,
                      const float* __restrict__ b2, float* __restrict__ wb) {
  __shared__ float f8[8], h[49];
  int r = blockIdx.x, t = threadIdx.x;
  if (t < 8) f8[t] = feats[r * 8 + t];
  __syncthreads();
  if (t < 49) {
    float s = b1[t];
#pragma unroll
    for (int k = 0; k < 8; ++k) s += f8[k] * w1[t * 8 + k];
    h[t] = leaky(s);
  }
  __syncthreads();
  if (t < 305) {
    float s = b2[t];
    for (int k = 0; k < 49; ++k) s += h[k] * w2[t * 49 + k];
    wb[r * 305 + t] = s;
  }
}

// ---------------- fused predicted-filter conv: layer0 via f32 WMMA, layer1 via VALU/LDS ----
// layer0 implicit GEMM per group: D(16 oc x 16 px) += A(16 oc x 4 k) * B(4 k x 16 px)
__global__ void __launch_bounds__(256) k_filter(const float* __restrict__ x,
                                                const float* __restrict__ wb,
                                                float* __restrict__ fx1) {
  __shared__ float fs[16 * FS_STRIDE];
  __shared__ float w1s[145];
  const int g = blockIdx.z;
  const int y0 = blockIdx.y * 16, x0 = blockIdx.x * 16;
  const float* xg = x + (size_t)g * HWIMG;
  const float* wbg = wb + g * 305;
  const int tid = threadIdx.x;
  if (tid < 145) w1s[tid] = wbg[160 + tid];  // 144 weights + bias at wb[304]
  const int lane = tid & 31, wave = tid >> 5;
  const int half = lane >> 4, n16 = lane & 15;

  // hint: pull the x tile into cache (gfx1250 global_prefetch)
  {
    int yr = refl(y0 - 2 + (tid >> 4), HH);
    int xr = refl(x0 - 2, WW);
    __builtin_prefetch(&xg[yr * WW + xr], 0, 1);
  }

  // fx0 tile (18x18 incl. reflect halo), 16 channels, via V_WMMA_F32_16X16X4_F32
  for (int t = wave; t < 21; t += 8) {
    int pix = t * 16 + n16;
    if (pix > 323) pix = 323;  // clamp lanes of last partial tile (duplicates identical)
    int yy = pix / 18, xx = pix % 18;
    int yr = refl(y0 - 1 + yy, HH);
    int xr = refl(x0 - 1 + xx, WW);
    v8f c = {};
    const int m = n16;  // A row (output channel) for this lane
#pragma unroll
    for (int k0 = 0; k0 < 12; k0 += 4) {
      int kk = k0 + 2 * half;  // lanes 0-15 hold K=k0,k0+1; lanes 16-31 hold K=k0+2,k0+3
      v2f a, b;
      a.x = (kk < 9) ? wbg[m * 9 + kk] : 0.f;
      a.y = (kk + 1 < 9) ? wbg[m * 9 + kk + 1] : 0.f;
      float bx = 0.f, by = 0.f;
      if (kk < 9)
        bx = xg[refl(yr + kk / 3 - 1, HH) * WW + refl(xr + kk % 3 - 1, WW)];
      if (kk + 1 < 9)
        by = xg[refl(yr + (kk + 1) / 3 - 1, HH) * WW + refl(xr + (kk + 1) % 3 - 1, WW)];
      b.x = bx;
      b.y = by;
      c = __builtin_amdgcn_wmma_f32_16x16x4_f32(false, a, false, b, (short)0, c,
                                                false, false);
    }
#pragma unroll
    for (int v = 0; v < 8; ++v) {
      int mo = v + 8 * half;  // C/D layout: VGPR v holds rows v (lanes 0-15), v+8 (16-31)
      fs[mo * FS_STRIDE + pix] = leaky(c[v] + wbg[144 + mo]);
    }
  }
  __syncthreads();

  // layer1: 16->1 3x3 over the LDS fx0 tile
  const int py = tid >> 4, px = tid & 15;
  const int oy = y0 + py, ox = x0 + px;
  float s = w1s[144];
#pragma unroll
  for (int ch = 0; ch < 16; ++ch)
#pragma unroll
    for (int dy = 0; dy < 3; ++dy)
#pragma unroll
      for (int dx = 0; dx < 3; ++dx)
        s += w1s[ch * 9 + dy * 3 + dx] * fs[ch * FS_STRIDE + (py + dy) * 18 + (px + dx)];
  if (ox < WW) fx1[(size_t)g * HWIMG + oy * WW + ox] = leaky(s);
}

// ---------------- GPF conv1 batch-stats pass (no store; L2-resident recompute later) ----
__global__ void k_g1stats(const float* __restrict__ fx1, const float* __restrict__ w,
                          const float* __restrict__ bias, float* __restrict__ bsum,
                          float* __restrict__ bsq) {
  __shared__ float wS[150], bS[6], accS[6], accQ[6];
  int tid = threadIdx.x;
  if (tid < 150) wS[tid] = w[tid];
  if (tid < 6) { bS[tid] = bias[tid]; accS[tid] = 0.f; accQ[tid] = 0.f; }
  __syncthreads();
  int b = blockIdx.x / 306;
  int pix = (blockIdx.x % 306) * 256 + tid;
  int y = pix / WW, xx = pix % WW;
  const float* img = fx1 + (size_t)b * HWIMG;
  float vin[25];
#pragma unroll
  for (int dy = 0; dy < 5; ++dy)
#pragma unroll
    for (int dx = 0; dx < 5; ++dx)
      vin[dy * 5 + dx] = img[refl(y + dy - 2, HH) * WW + refl(xx + dx - 2, WW)];
#pragma unroll
  for (int c = 0; c < 6; ++c) {
    float s = bS[c];
#pragma unroll
    for (int j = 0; j < 25; ++j) s += wS[c * 25 + j] * vin[j];
    float rs = waveRed(s), rq = waveRed(s * s);
    if ((tid & 31) == 0) { atomicAdd(&accS[c], rs); atomicAdd(&accQ[c], rq); }
  }
  __syncthreads();
  if (tid < 6) { atomicAdd(&bsum[tid], accS[tid]); atomicAdd(&bsq[tid], accQ[tid]); }
}

__global__ void k_bnfin(const float* bsum, const float* bsq, const float* g,
                        const float* b, float* scale, float* shift, int nch, float cnt) {
  int c = threadIdx.x;
  if (c >= nch) return;
  float mu = bsum[c] / cnt;
  float var = bsq[c] / cnt - mu * mu;
  float r = rsqrtf(var + 1e-5f);
  scale[c] = r * g[c];
  shift[c] = b[c] - mu * r * g[c];
}

// ---------------- fused GPF: conv1+BN1+ReLU tile (LDS) -> conv2 -> y2 (+BN2 stats) ------
__global__ void __launch_bounds__(256) k_g2(const float* __restrict__ fx1,
                                            const float* __restrict__ w1,
                                            const float* __restrict__ b1,
                                            const float* __restrict__ bn1sc,
                                            const float* __restrict__ bn1sh,
                                            const float* __restrict__ w2,
                                            const float* __restrict__ b2,
                                            float* __restrict__ y2, float* __restrict__ bsum,
                                            float* __restrict__ bsq) {
  __shared__ float g1s[6 * G1S];
  __shared__ float w1S[150], w2S[150], b1S[6], s1S[6], h1S[6];
  __shared__ float accS, accQ;
  int tid = threadIdx.x;
  if (tid < 150) { w1S[tid] = w1[tid]; w2S[tid] = w2[tid]; }
  if (tid < 6) { b1S[tid] = b1[tid]; s1S[tid] = bn1sc[tid]; h1S[tid] = bn1sh[tid]; }
  if (tid == 0) { accS = 0.f; accQ = 0.f; }
  __syncthreads();
  int b = blockIdx.z;
  int y0 = blockIdx.y * 16, x0 = blockIdx.x * 16;
  const float* img = fx1 + (size_t)b * HWIMG;
  for (int idx = tid; idx < 2400; idx += 256) {
    int c = idx / 400, q = idx % 400;
    int yy = q / 20, xx = q % 20;
    int yr = refl(y0 - 2 + yy, HH), xr = refl(x0 - 2 + xx, WW);
    float s = b1S[c];
#pragma unroll
    for (int dy = 0; dy < 5; ++dy)
#pragma unroll
      for (int dx = 0; dx < 5; ++dx)
        s += w1S[c * 25 + dy * 5 + dx] *
             img[refl(yr + dy - 2, HH) * WW + refl(xr + dx - 2, WW)];
    g1s[c * G1S + q] = fmaxf(s * s1S[c] + h1S[c], 0.f);
  }
  __syncthreads();
  int py = tid >> 4, px = tid & 15;
  int oy = y0 + py, ox = x0 + px;
  float s = b2[0];
#pragma unroll
  for (int c = 0; c < 6; ++c)
#pragma unroll
    for (int dy = 0; dy < 5; ++dy)
#pragma unroll
      for (int dx = 0; dx < 5; ++dx)
        s += w2S[c * 25 + dy * 5 + dx] * g1s[c * G1S + (py + dy) * 20 + (px + dx)];
  bool valid = ox < WW;
  if (valid) y2[(size_t)b * HWIMG + oy * WW + ox] = s;
  float sv = valid ? s : 0.f;
  float rs = waveRed(sv), rq = waveRed(sv * sv);
  if ((tid & 31) == 0) { atomicAdd(&accS, rs); atomicAdd(&accQ, rq); }
  __syncthreads();
  if (tid == 0) { atomicAdd(&bsum[0], accS); atomicAdd(&bsq[0], accQ); }
}

// ---------------- final combine ----------------
__global__ void k_final(const float* __restrict__ fx1, const float* __restrict__ y2,
                        const float* __restrict__ bn2sc, const float* __restrict__ bn2sh,
                        const float* __restrict__ cw, const float* __restrict__ cb,
                        float* __restrict__ out) {
  size_t i = (size_t)blockIdx.x * blockDim.x + threadIdx.x;
  float g2v = fmaxf(y2[i] * bn2sc[0] + bn2sh[0], 0.f);
  out[i] = cw[0] * fx1[i] + cw[1] * g2v + cb[0];
}

extern "C" void kernel_launch(void* const* d_in, const int* in_sizes, int n_in,
                              void* d_out, int out_size, void* d_ws, size_t ws_size,
                              hipStream_t stream) {
  (void)in_sizes; (void)n_in; (void)out_size; (void)ws_size;
  const float* x      = (const float*)d_in[0];
  const float* enc_w1 = (const float*)d_in[1];
  const float* enc_b1 = (const float*)d_in[2];
  const float* gn1_g  = (const float*)d_in[3];
  const float* gn1_b  = (const float*)d_in[4];
  const float* enc_w2 = (const float*)d_in[5];
  const float* enc_b2 = (const float*)d_in[6];
  const float* gn2_g  = (const float*)d_in[7];
  const float* gn2_b  = (const float*)d_in[8];
  const float* enc_w3 = (const float*)d_in[9];
  const float* enc_b3 = (const float*)d_in[10];
  const float* gn3_g  = (const float*)d_in[11];
  const float* gn3_b  = (const float*)d_in[12];
  const float* lin1_w = (const float*)d_in[13];
  const float* lin1_b = (const float*)d_in[14];
  const float* lin2_w = (const float*)d_in[15];
  const float* lin2_b = (const float*)d_in[16];
  const float* gpf_w1 = (const float*)d_in[17];
  const float* gpf_b1 = (const float*)d_in[18];
  const float* bn1_g  = (const float*)d_in[19];
  const float* bn1_b  = (const float*)d_in[20];
  const float* gpf_w2 = (const float*)d_in[21];
  const float* gpf_b2 = (const float*)d_in[22];
  const float* bn2_g  = (const float*)d_in[23];
  const float* bn2_b  = (const float*)d_in[24];
  const float* comb_w = (const float*)d_in[25];
  const float* comb_b = (const float*)d_in[26];

  float* ws   = (float*)d_ws;
  float* fx1  = ws;                   // 64*192*408       = 5,013,504
  float* buf1 = ws + 5013504;         // e1pre (2*64*190*203 = 4,936,960); reused as y2
  float* buf2 = buf1 + 4935040;       // e2pre (2*128*188*100 = 4,812,800)
  float* wb   = buf2 + 4812800;       // 64*305 = 19,520
  float* sm   = wb + 19520;           // small stats area (2112 floats)
  float* gn1sum = sm;
  float* gn1sq  = sm + 64;
  float* gn2sum = sm + 128;
  float* gn2sq  = sm + 192;
  float* scA    = sm + 256;           // 128
  float* shA    = sm + 384;           // 128
  float* scB    = sm + 512;           // 256
  float* shB    = sm + 768;           // 256
  float* pooled = sm + 1024;          // 512
  float* feats  = sm + 1536;          // 512
  float* bn1sum = sm + 2048;
  float* bn1sq  = sm + 2056;
  float* bn1sc  = sm + 2064;
  float* bn1sh  = sm + 2072;
  float* bn2sum = sm + 2080;
  float* bn2sq  = sm + 2084;
  float* bn2sc  = sm + 2088;
  float* bn2sh  = sm + 2092;
  float* y2     = buf1;               // encoder buffers dead by GPF phase
  float* out    = (float*)d_out;

  k_zero<<<(2112 + 255) / 256, 256, 0, stream>>>(sm, 2112);
  k_enc1<<<128, 256, 0, stream>>>(x, enc_w1, enc_b1, buf1, gn1sum, gn1sq);
  k_gnfin<<<1, 128, 0, stream>>>(gn1sum, gn1sq, gn1_g, gn1_b, scA, shA, 64, 2,
                                 2.f * 190.f * 203.f);
  k_enc2<<<256, 256, 0, stream>>>(buf1, enc_w2, enc_b2, scA, shA, buf2, gn2sum, gn2sq);
  k_gnfin<<<1, 256, 0, stream>>>(gn2sum, gn2sq, gn2_g, gn2_b, scB, shB, 128, 4,
                                 4.f * 188.f * 100.f);
  k_enc3<<<512, 256, 0, stream>>>(buf2, enc_w3, enc_b3, scB, shB, pooled);
  k_gn3feats<<<1, 512, 0, stream>>>(pooled, gn3_g, gn3_b, feats);
  k_mlp<<<64, 320, 0, stream>>>(feats, lin1_w, lin1_b, lin2_w, lin2_b, wb);
  dim3 gf(26, 12, 64);
  k_filter<<<gf, 256, 0, stream>>>(x, wb, fx1);
  k_g1stats<<<64 * 306, 256, 0, stream>>>(fx1, gpf_w1, gpf_b1, bn1sum, bn1sq);
  k_bnfin<<<1, 32, 0, stream>>>(bn1sum, bn1sq, bn1_g, bn1_b, bn1sc, bn1sh, 6,
                                64.f * (float)HWIMG);
  k_g2<<<gf, 256, 0, stream>>>(fx1, gpf_w1, gpf_b1, bn1sc, bn1sh, gpf_w2, gpf_b2, y2,
                               bn2sum, bn2sq);
  k_bnfin<<<1, 32, 0, stream>>>(bn2sum, bn2sq, bn2_g, bn2_b, bn2sc, bn2sh, 1,
                                64.f * (float)HWIMG);
  k_final<<<64 * 306, 256, 0, stream>>>(fx1, y2, bn2sc, bn2sh, comb_w, comb_b, out);
}